// EfficientDet_68470368632937
// MI455X (gfx1250) — compile-verified
//
#include <hip/hip_runtime.h>
#include <hip/hip_bf16.h>

#define BATCH 8
#define NA 49104
#define NC 80
#define TOPK 256
#define IMGF 512.0f
#define THRESH 0.01f
#define IOU_THR 0.5f
#define NBINS 4096

typedef float v2f __attribute__((ext_vector_type(2)));
typedef float v8f __attribute__((ext_vector_type(8)));

// ---------------- Kernel 1: sigmoid-max + argmax + box decode (memory-bound stream) -------------
__global__ void k_score_decode(const float* __restrict__ cls,
                               const float* __restrict__ regs,
                               const float* __restrict__ anchors,
                               float* __restrict__ wsBoxes,
                               float* __restrict__ wsScores,
                               int* __restrict__ wsLabels,
                               int total) {
  int gid = blockIdx.x * blockDim.x + threadIdx.x;
  if (gid >= total) return;
  int a = gid % NA;

  // prefetch the small side streams while the 320B logits row is in flight
  __builtin_prefetch(regs + (size_t)gid * 4, 0, 3);
  __builtin_prefetch(anchors + (size_t)a * 4, 0, 3);

  const float4* row = (const float4*)(cls + (size_t)gid * NC);
  float m = -3.0e38f;
  int mi = 0;
#pragma unroll
  for (int q = 0; q < NC / 4; ++q) {
    float4 v = row[q];
    if (v.x > m) { m = v.x; mi = q * 4 + 0; }
    if (v.y > m) { m = v.y; mi = q * 4 + 1; }
    if (v.z > m) { m = v.z; mi = q * 4 + 2; }
    if (v.w > m) { m = v.w; mi = q * 4 + 3; }
  }
  // max(sigmoid(x)) == sigmoid(max(x)); argmax identical (monotone)
  float s = 1.0f / (1.0f + __expf(-m));
  if (!(s > THRESH)) s = 0.0f;
  wsScores[gid] = s;
  wsLabels[gid] = mi;

  float4 an = ((const float4*)anchors)[a];
  float4 rg = ((const float4*)regs)[gid];
  float aw = an.z - an.x, ah = an.w - an.y;
  float acx = an.x + 0.5f * aw, acy = an.y + 0.5f * ah;
  float cx = acx + rg.x * aw;
  float cy = acy + rg.y * ah;
  float w = aw * __expf(fminf(fmaxf(rg.z, -4.0f), 4.0f));
  float h = ah * __expf(fminf(fmaxf(rg.w, -4.0f), 4.0f));
  float4 bx;
  bx.x = fminf(fmaxf(cx - 0.5f * w, 0.0f), IMGF);
  bx.y = fminf(fmaxf(cy - 0.5f * h, 0.0f), IMGF);
  bx.z = fminf(fmaxf(cx + 0.5f * w, 0.0f), IMGF);
  bx.w = fminf(fmaxf(cy + 0.5f * h, 0.0f), IMGF);
  ((float4*)wsBoxes)[gid] = bx;
}

// ---------------- Kernel 2: per-batch top-256 via radix histogram + bitonic sort ----------------
__global__ __launch_bounds__(1024)
void k_topk(const float* __restrict__ wsScores,
            const float* __restrict__ wsBoxes,
            const int* __restrict__ wsLabels,
            float* __restrict__ topB,
            float* __restrict__ topS,
            float* __restrict__ out) {
  const int b = blockIdx.x;
  const int tid = threadIdx.x;
  const int NT = 1024;

  __shared__ unsigned hist[NBINS];
  __shared__ unsigned chunkSum[64];
  __shared__ float sKey[TOPK];
  __shared__ int sVal[TOPK];
  __shared__ unsigned s_cut, s_nab, s_need, s_cA, s_cB;

  for (int i = tid; i < NBINS; i += NT) hist[i] = 0;
  __syncthreads();

  const float* sc = wsScores + (size_t)b * NA;
  for (int i = tid; i < NA; i += NT) {
    float s = sc[i];
    if (s > 0.0f) atomicAdd(&hist[__float_as_uint(s) >> 19], 1u);
  }
  __syncthreads();

  if (tid < 64) {
    unsigned acc = 0;
#pragma unroll 8
    for (int q = 0; q < 64; ++q) acc += hist[tid * 64 + q];
    chunkSum[tid] = acc;
  }
  __syncthreads();

  if (tid == 0) {
    unsigned cum = 0;
    int cut = -1;
    unsigned nab = 0;
    for (int c = 63; c >= 0; --c) {
      unsigned cs = chunkSum[c];
      if (cum + cs >= TOPK) {
        for (int k = c * 64 + 63; k >= c * 64; --k) {
          unsigned h = hist[k];
          if (cum + h >= TOPK) { cut = k; nab = cum; break; }
          cum += h;
        }
        break;
      }
      cum += cs;
    }
    if (cut < 0) { cut = 0; nab = 0; }  // < TOPK positives: all keys (>=1) take the '>' path
    s_cut = (unsigned)cut;
    s_nab = nab;
    s_need = TOPK - nab;
    s_cA = 0;
    s_cB = 0;
  }
  __syncthreads();
  for (int i = tid; i < TOPK; i += NT) { sKey[i] = 0.0f; sVal[i] = 0; }
  __syncthreads();

  unsigned cut = s_cut, nab = s_nab, need = s_need;
  for (int i = tid; i < NA; i += NT) {
    float s = sc[i];
    if (s <= 0.0f) continue;
    unsigned k = __float_as_uint(s) >> 19;
    if (k > cut) {
      unsigned p = atomicAdd(&s_cA, 1u);
      if (p < TOPK) { sKey[p] = s; sVal[p] = i; }
    } else if (k == cut) {
      unsigned p = atomicAdd(&s_cB, 1u);
      if (p < need) { sKey[nab + p] = s; sVal[nab + p] = i; }
    }
  }
  __syncthreads();

  // bitonic sort, descending by score (NMS consumes score-sorted order)
  for (unsigned k2 = 2; k2 <= TOPK; k2 <<= 1) {
    for (unsigned j = k2 >> 1; j > 0; j >>= 1) {
      if (tid < TOPK) {
        unsigned ixj = tid ^ j;
        if (ixj > tid) {
          float a = sKey[tid], c = sKey[ixj];
          bool dosw = ((tid & k2) == 0) ? (a < c) : (a > c);
          if (dosw) {
            sKey[tid] = c; sKey[ixj] = a;
            int t = sVal[tid]; sVal[tid] = sVal[ixj]; sVal[ixj] = t;
          }
        }
      }
      __syncthreads();
    }
  }

  if (tid < TOPK) {
    int idx = sVal[tid];
    float s = sKey[tid];
    size_t base = (size_t)b * NA + (size_t)idx;
    float4 bx = ((const float4*)wsBoxes)[base];
    ((float4*)topB)[(size_t)b * TOPK + tid] = bx;
    topS[(size_t)b * TOPK + tid] = s;
    // labels region of d_out: offset BATCH*TOPK*5
    out[(size_t)BATCH * TOPK * 5 + (size_t)b * TOPK + tid] = (float)wsLabels[base];
  }
}

// ---------------- Kernel 3: NMS. Pairwise union rank-2 term via V_WMMA_F32_16X16X4_F32 ----------
__global__ __launch_bounds__(256)
void k_nms(const float* __restrict__ topB,
           const float* __restrict__ topS,
           float* __restrict__ out) {
  const int b = blockIdx.x;
  const int tid = threadIdx.x;  // 256 threads = 8 waves (wave32)
  __shared__ float bx1[TOPK], by1[TOPK], bx2[TOPK], by2[TOPK], sar[TOPK], ssc[TOPK];
  __shared__ unsigned rowmask[TOPK][8];  // bit i of rowmask[j]: iou(j,i) > 0.5
  __shared__ unsigned keepArr[TOPK];

  float4 box = ((const float4*)topB)[(size_t)b * TOPK + tid];
  float s = topS[(size_t)b * TOPK + tid];
  bx1[tid] = box.x; by1[tid] = box.y; bx2[tid] = box.z; by2[tid] = box.w;
  float area = fmaxf(box.z - box.x, 0.0f) * fmaxf(box.w - box.y, 0.0f);
  sar[tid] = area;
  ssc[tid] = s;
#pragma unroll
  for (int q = 0; q < 8; ++q) rowmask[tid][q] = 0;
  __syncthreads();

  const int wave = tid >> 5;
  const int lane = tid & 31;

  // 16x16 tiles over the (j,i) pair grid: 256 tiles, 32 per wave. Loop is wave-uniform
  // so EXEC is all-ones at every WMMA (ISA 7.12 requirement).
  for (int t = wave; t < 256; t += 8) {
    const int j0 = (t >> 4) << 4;
    const int i0 = (t & 15) << 4;
    // A (16x4 f32): row m = [area_{j0+m}, 1, 0, 0]; lanes 0-15 hold K=0,1; lanes 16-31 K=2,3 (zero)
    // B (4x16 f32): col n = [1, area_{i0+n}, 0, 0]^T
    v2f Aop, Bop;
    if (lane < 16) {
      Aop.x = sar[j0 + lane]; Aop.y = 1.0f;
      Bop.x = 1.0f;           Bop.y = sar[i0 + lane];
    } else {
      Aop.x = 0.0f; Aop.y = 0.0f;
      Bop.x = 0.0f; Bop.y = 0.0f;
    }
    v8f Cz = {0.0f, 0.0f, 0.0f, 0.0f, 0.0f, 0.0f, 0.0f, 0.0f};
    // D[m][n] = area_{j0+m} + area_{i0+n}  (rank-2 union term on the matrix pipe)
    v8f D = __builtin_amdgcn_wmma_f32_16x16x4_f32(false, Aop, false, Bop,
                                                  (short)0, Cz, false, false);
    const int i = i0 + (lane & 15);
    const int jbase = j0 + ((lane >> 4) << 3);  // D lane layout: M = r + 8*(lane/16)
    float ix1 = bx1[i], iy1 = by1[i], ix2 = bx2[i], iy2 = by2[i];
#pragma unroll
    for (int r = 0; r < 8; ++r) {
      int j = jbase + r;
      float ltx = fmaxf(bx1[j], ix1), lty = fmaxf(by1[j], iy1);
      float rbx = fminf(bx2[j], ix2), rby = fminf(by2[j], iy2);
      float inter = fmaxf(rbx - ltx, 0.0f) * fmaxf(rby - lty, 0.0f);
      float uni = fmaxf(D[r] - inter, 1e-8f);
      bool hit = inter > IOU_THR * uni;
      unsigned bal = __builtin_amdgcn_ballot_w32(hit);
      if (lane == 0) {
        unsigned lo = bal & 0xFFFFu, hi = bal >> 16;
        int word = i0 >> 5, sh = i0 & 16;
        if (lo) atomicOr(&rowmask[j0 + r][word], lo << sh);
        if (hi) atomicOr(&rowmask[j0 + 8 + r][word], hi << sh);
      }
    }
  }
  __syncthreads();

  // sequential suppression scan; wave 0 handles 8 rows per lane in parallel
  if (wave == 0) {
    bool kp[8];
#pragma unroll
    for (int r = 0; r < 8; ++r) kp[r] = ssc[lane * 8 + r] > 0.0f;
    for (int i = 0; i < 256; ++i) {
      bool loc = false;
#pragma unroll
      for (int r = 0; r < 8; ++r) {
        int j = lane * 8 + r;
        if (j < i && kp[r] && ((rowmask[j][i >> 5] >> (i & 31)) & 1u)) loc = true;
      }
      unsigned sup = __builtin_amdgcn_ballot_w32(loc);
      if (sup && lane == (i >> 3)) kp[i & 7] = false;
    }
#pragma unroll
    for (int r = 0; r < 8; ++r) keepArr[lane * 8 + r] = kp[r] ? 1u : 0u;
  }
  __syncthreads();

  float k = keepArr[tid] ? 1.0f : 0.0f;
  size_t db = ((size_t)b * TOPK + tid) * 5;
  out[db + 0] = box.x * k;
  out[db + 1] = box.y * k;
  out[db + 2] = box.z * k;
  out[db + 3] = box.w * k;
  out[db + 4] = s * k;
  out[(size_t)BATCH * TOPK * 6 + (size_t)b * TOPK + tid] = k;  // keep region
}

// ---------------- launch -----------------------------------------------------------------------
extern "C" void kernel_launch(void* const* d_in, const int* in_sizes, int n_in,
                              void* d_out, int out_size, void* d_ws, size_t ws_size,
                              hipStream_t stream) {
  const float* cls = (const float*)d_in[0];      // [8,49104,80]
  const float* regs = (const float*)d_in[1];     // [8,49104,4]
  const float* anchors = (const float*)d_in[2];  // [49104,4]
  float* out = (float*)d_out;                    // dets(10240) + labels(2048) + keep(2048)
  float* ws = (float*)d_ws;

  const size_t BA = (size_t)BATCH * NA;
  float* wsBoxes  = ws;                      // BA*4 floats
  float* wsScores = ws + BA * 4;             // BA floats
  int*   wsLabels = (int*)(ws + BA * 5);     // BA ints
  float* wsTopB   = ws + BA * 6;             // BATCH*TOPK*4 floats
  float* wsTopS   = wsTopB + BATCH * TOPK * 4;

  int total = (int)BA;
  k_score_decode<<<(total + 255) / 256, 256, 0, stream>>>(cls, regs, anchors,
                                                          wsBoxes, wsScores, wsLabels, total);
  k_topk<<<BATCH, 1024, 0, stream>>>(wsScores, wsBoxes, wsLabels, wsTopB, wsTopS, out);
  k_nms<<<BATCH, 256, 0, stream>>>(wsTopB, wsTopS, out);
}